// FCNNColorValuationFunction_39977555591631
// MI455X (gfx1250) — compile-verified
//
#include <hip/hip_runtime.h>
#include <stdint.h>

// FCNNColorValuationFunction: out[row] = a[row, argmax(z[row,3:6])]
// Streaming kernel: 48 B/row * 4.19M rows ~= 201 MB -> ~8.6 us floor at 23.3 TB/s.

typedef float v4f __attribute__((ext_vector_type(4)));

#define THREADS 256
#define ROWS_PER_THREAD 4
#define ROWS_PER_BLOCK (THREADS * ROWS_PER_THREAD)   // 1024 rows
#define A_TILE_FLOATS (ROWS_PER_BLOCK * 3)           // 3072 floats = 12 KB

__global__ __launch_bounds__(THREADS)
void color_valuation_kernel(const float* __restrict__ z,
                            const float* __restrict__ a,
                            float* __restrict__ out,
                            long long n_rows)
{
    __shared__ float a_tile[A_TILE_FLOATS];   // 12 KB of 320 KB/WGP

    const int       t        = threadIdx.x;
    const long long row0_blk = (long long)blockIdx.x * ROWS_PER_BLOCK;
    const long long row0     = row0_blk + (long long)t * ROWS_PER_THREAD;
    const v4f*      zq       = (const v4f*)z;   // 2 x v4f per 8-float z row

    if (row0_blk + ROWS_PER_BLOCK <= n_rows) {
        // ---------------- full block: async-stage a-tile into LDS ----------------
        // chunk c = t + k*256, 16 B each: lanes stride 16 B -> perfectly coalesced,
        // 768 b128 transfers = 12288 B per block.
        const char* ga   = (const char*)a + row0_blk * 3 * sizeof(float);
        unsigned    lds0 = (unsigned)(uintptr_t)(&a_tile[0]);
        #pragma unroll
        for (int k = 0; k < 3; ++k) {
            unsigned    off  = (unsigned)(t + k * THREADS) * 16u;
            const char* gsrc = ga + off;
            unsigned    ldst = lds0 + off;
            asm volatile("global_load_async_to_lds_b128 %0, %1, off"
                         :: "v"(ldst), "v"(gsrc)
                         : "memory");
        }
#if __has_builtin(__builtin_amdgcn_s_wait_asynccnt)
        __builtin_amdgcn_s_wait_asynccnt(0);
#else
        asm volatile("s_wait_asynccnt 0" ::: "memory");
#endif
        __syncthreads();   // chunks interleave across waves -> block barrier

        const float* my_a = &a_tile[t * (ROWS_PER_THREAD * 3)];  // 12 floats/thread
        v4f res;
        #pragma unroll
        for (int r = 0; r < ROWS_PER_THREAD; ++r) {
            long long row = row0 + r;
            // need z[3] (lo.w), z[4],z[5] (hi.x,hi.y); NT: read-once stream
            v4f lo = __builtin_nontemporal_load(zq + row * 2);
            v4f hi = __builtin_nontemporal_load(zq + row * 2 + 1);
            float c0 = lo.w, c1 = hi.x, c2 = hi.y;
            // first-occurrence argmax (jnp.argmax tie-break)
            int idx = (c0 >= c1 && c0 >= c2) ? 0 : ((c1 >= c2) ? 1 : 2);
            res[r] = my_a[r * 3 + idx];                   // ds_load_b32 gather
        }
        // 4 results -> one coalesced NT b128 store
        __builtin_nontemporal_store(res, (v4f*)out + (row0 >> 2));
        return;
    }

    // ---------------- tail block (unused for B = 4M): direct scalar path --------
    for (int r = 0; r < ROWS_PER_THREAD; ++r) {
        long long row = row0 + r;
        if (row >= n_rows) return;
        float c0 = z[row * 8 + 3];
        float c1 = z[row * 8 + 4];
        float c2 = z[row * 8 + 5];
        int idx = (c0 >= c1 && c0 >= c2) ? 0 : ((c1 >= c2) ? 1 : 2);
        out[row] = a[row * 3 + idx];
    }
}

extern "C" void kernel_launch(void* const* d_in, const int* in_sizes, int n_in,
                              void* d_out, int out_size, void* d_ws, size_t ws_size,
                              hipStream_t stream)
{
    const float* z = (const float*)d_in[0];   // (B, 8) float32
    const float* a = (const float*)d_in[1];   // (B, 3) float32
    float* out     = (float*)d_out;           // (B,)  float32

    long long n_rows = (long long)in_sizes[0] / 8;
    int blocks = (int)((n_rows + ROWS_PER_BLOCK - 1) / ROWS_PER_BLOCK);

    color_valuation_kernel<<<dim3(blocks), dim3(THREADS), 0, stream>>>(z, a, out, n_rows);
}